// VulGAT_49039936585887
// MI455X (gfx1250) — compile-verified
//
#include <hip/hip_runtime.h>
#include <math.h>

// ---------------------------------------------------------------------------
// VulGAT forward for MI455X (gfx1250, wave32).
//
// Input pointer layout assumption (setup_inputs dict insertion order, params
// flattened as a JAX pytree with alphabetically sorted keys):
//   0: x[50000,128]            1: edge_attr[400000,16]
//   conv0: 2:We[16,256] 3:Wl[128,256] 4:Wr[128,256] 5:att[8,32] 6:bias[256] 7:bl[256] 8:br[256]
//   conv1: 9:We 10:Wl[256,256] 11:Wr[256,256] 12:att 13:bias 14:bl 15:br
//   lin0: 16:W[512,256] 17:b[256]   lin1: 18:W[256,128] 19:b[128]
//   lin_out: 20:W[128,2] 21:b[2]
//   norm0: 22:alpha 23:bias 24:weight   norm1: 25:alpha 26:bias 27:weight
//   pool0: 28:w[256]  pool1: 29:w[256]
//   30: edge_index[2,400000] (int32)   31: batch[50000] (int32)
// Output: log-softmax logits [50,2] float32.
// ---------------------------------------------------------------------------

#define NN   50000
#define NG   50
#define NPG  1000
#define EE   400000
#define FE   16
#define HH   8
#define CC   32
#define DD   256
#define NEG_INF (-__builtin_inff())

typedef __attribute__((ext_vector_type(2))) float v2f;
typedef __attribute__((ext_vector_type(8))) float v8f;

#if __has_builtin(__builtin_amdgcn_global_load_async_to_lds_b32)
#define ASYNC_STAGE 1
typedef __attribute__((address_space(1))) int* gptr_i32;
typedef __attribute__((address_space(3))) int* lptr_i32;
#endif

static __device__ __forceinline__ unsigned fenc(float f) {
  // order-preserving float -> uint map (for atomicMax-based segment max)
  unsigned u = __float_as_uint(f);
  return u ^ (unsigned)(((int)u >> 31) | 0x80000000);
}
static __device__ __forceinline__ float fdec(unsigned u) {
  unsigned v = (u & 0x80000000u) ? (u ^ 0x80000000u) : ~u;
  return __uint_as_float(v);
}

// --------------------------- utility kernels --------------------------------

__global__ void k_init_masks(int* __restrict__ nmask, int* __restrict__ emask) {
  int t = blockIdx.x * blockDim.x + threadIdx.x;
  if (t < NN) nmask[t] = 1;
  if (t < EE) emask[t] = 1;
}

__global__ void k_zero(float* __restrict__ p, int n) {
  int t = blockIdx.x * blockDim.x + threadIdx.x;
  if (t < n) p[t] = 0.0f;
}

// --------------------------- WMMA fp32 GEMM ---------------------------------
// Y[N,OUT] = X[N,K] @ W[K,OUT] + bias, exact fp32 via V_WMMA_F32_16X16X4_F32.
// One workgroup = 8 waves sharing ONE 16-column tile of W staged in LDS
// (<=16 KB of the WGP's 320 KB); each wave computes a different 16-row tile.
// This cuts B-matrix L2 traffic 8x vs per-wave fetch and turns the strided
// B reads into conflict-free ds_load_b32 (the two K-halves per wave are 32
// dwords apart -> disjoint bank ranges for lanes 0-15 vs 16-31).
// Staging uses GLOBAL_LOAD_ASYNC_TO_LDS_B32 (ASYNCcnt) when the builtin is
// available, else plain LDS stores.
// Lane layouts per CDNA5 ISA 7.12.2:
//   A 16x4 f32 : lane m=L&15; VGPR0 holds K=(L>>4)*2, VGPR1 holds K+1
//   B 4x16 f32 : lane n=L&15; VGPR0 holds K=(L>>4)*2, VGPR1 holds K+1
//   C/D 16x16  : VGPR r -> row r (lanes 0-15) / row r+8 (lanes 16-31), col=L&15
__global__ __launch_bounds__(256) void k_wmma_gemm_bias(
    const float* __restrict__ X, const float* __restrict__ W,
    const float* __restrict__ bias, float* __restrict__ Y,
    int N, int K, int OUT) {
  __shared__ float Bs[256 * 16];  // max K=256

  const int lane = threadIdx.x & 31;
  const int wave = threadIdx.x >> 5;
  const int tn = blockIdx.y << 4;               // column tile (shared by block)
  const int tm = (blockIdx.x * 8 + wave) << 4;  // row tile (per wave)

  // ---- stage B tile W[0:K, tn:tn+16] into LDS (all threads participate) ----
  for (int i = threadIdx.x; i < K * 16; i += 256) {
    const int kk = i >> 4, n = i & 15;
#ifdef ASYNC_STAGE
    __builtin_amdgcn_global_load_async_to_lds_b32(
        (gptr_i32)(W + (size_t)kk * OUT + tn + n), (lptr_i32)&Bs[i], 0, 0);
#else
    Bs[i] = W[(size_t)kk * OUT + tn + n];
#endif
  }
#ifdef ASYNC_STAGE
#if __has_builtin(__builtin_amdgcn_s_wait_asynccnt)
  __builtin_amdgcn_s_wait_asynccnt(0);
#else
  asm volatile("s_wait_asynccnt 0x0" ::: "memory");
#endif
#endif
  __syncthreads();

  if (tm >= N) return;  // wave-uniform tail exit: EXEC all-ones for WMMA

  const int mn = lane & 15;          // A row / B col for this lane
  const int kh = (lane >> 4) << 1;   // 0 or 2 (K sub-offset per half-wave)
  const float* __restrict__ xrow = X + (size_t)(tm + mn) * K;

  v8f acc = {};
  for (int k = 0; k < K; k += 4) {
    __builtin_prefetch(xrow + k + 32, 0, 1);  // -> global_prefetch_b8
    v2f a, b;
    a.x = xrow[k + kh];
    a.y = xrow[k + kh + 1];
    b.x = Bs[(k + kh) * 16 + mn];
    b.y = Bs[(k + kh + 1) * 16 + mn];
    acc = __builtin_amdgcn_wmma_f32_16x16x4_f32(
        /*neg_a=*/false, a, /*neg_b=*/false, b,
        /*c_mod=*/(short)0, acc, /*reuse_a=*/false, /*reuse_b=*/false);
  }
  const int rowoff = (lane >> 4) << 3;  // lanes 16-31 hold rows 8..15
  const int col = tn + mn;
  const float bb = bias[col];
#pragma unroll
  for (int r = 0; r < 8; ++r) {
    Y[(size_t)(tm + r + rowoff) * OUT + col] = acc[r] + bb;
  }
}

// ----------------------- GATv2 attention pipeline ---------------------------

// Accumulate mean of active incoming edge_attr per dst node (self-loop attr).
__global__ void k_selfattr(const float* __restrict__ eattr,
                           const int* __restrict__ src, const int* __restrict__ dst,
                           const int* __restrict__ emask,
                           float* __restrict__ sacc, float* __restrict__ scnt) {
  int e = blockIdx.x * blockDim.x + threadIdx.x;
  if (e >= EE) return;
  int s = src[e], d = dst[e];
  if (!emask[e] || s == d) return;
#pragma unroll
  for (int f = 0; f < FE; ++f)
    atomicAdd(&sacc[(size_t)d * FE + f], eattr[(size_t)e * FE + f]);
  atomicAdd(&scnt[d], 1.0f);
}

// e_edge[e,h] = sum_c leaky_relu(xl[src]+xr[dst]+eattr@We, 0.2) * att ; -inf if masked
__global__ void k_edge_scores(const float* __restrict__ xl, const float* __restrict__ xr,
                              const float* __restrict__ eattr, const float* __restrict__ We,
                              const float* __restrict__ att,
                              const int* __restrict__ src, const int* __restrict__ dst,
                              const int* __restrict__ emask, float* __restrict__ eedge) {
  int t = blockIdx.x * blockDim.x + threadIdx.x;
  if (t >= EE * HH) return;
  int e = t >> 3, h = t & 7;
  int s = src[e], d = dst[e];
  if (!emask[e] || s == d) { eedge[t] = NEG_INF; return; }
  float ea[FE];
#pragma unroll
  for (int f = 0; f < FE; ++f) ea[f] = eattr[(size_t)e * FE + f];
  const float* lp = xl + (size_t)s * DD + h * CC;
  const float* rp = xr + (size_t)d * DD + h * CC;
  const float* ap = att + h * CC;
  float acc = 0.0f;
  for (int c = 0; c < CC; ++c) {
    float v = 0.0f;
#pragma unroll
    for (int f = 0; f < FE; ++f) v += ea[f] * We[f * DD + h * CC + c];
    v += lp[c] + rp[c];
    v = v > 0.0f ? v : 0.2f * v;
    acc += v * ap[c];
  }
  eedge[t] = acc;
}

// e_self[n,h] from self-loop (mean edge attr); also seed segment max m.
__global__ void k_self_scores(const float* __restrict__ xl, const float* __restrict__ xr,
                              const float* __restrict__ We, const float* __restrict__ att,
                              const float* __restrict__ sacc, const float* __restrict__ scnt,
                              float* __restrict__ eself, unsigned* __restrict__ menc) {
  int t = blockIdx.x * blockDim.x + threadIdx.x;
  if (t >= NN * HH) return;
  int n = t >> 3, h = t & 7;
  float cnt = fmaxf(scnt[n], 1.0f);
  float sa[FE];
#pragma unroll
  for (int f = 0; f < FE; ++f) sa[f] = sacc[(size_t)n * FE + f] / cnt;
  const float* lp = xl + (size_t)n * DD + h * CC;
  const float* rp = xr + (size_t)n * DD + h * CC;
  const float* ap = att + h * CC;
  float acc = 0.0f;
  for (int c = 0; c < CC; ++c) {
    float v = 0.0f;
#pragma unroll
    for (int f = 0; f < FE; ++f) v += sa[f] * We[f * DD + h * CC + c];
    v += lp[c] + rp[c];
    v = v > 0.0f ? v : 0.2f * v;
    acc += v * ap[c];
  }
  eself[t] = acc;
  menc[t] = fenc(acc);
}

__global__ void k_edge_max(const float* __restrict__ eedge, const int* __restrict__ dst,
                           unsigned* __restrict__ menc) {
  int t = blockIdx.x * blockDim.x + threadIdx.x;
  if (t >= EE * HH) return;
  float v = eedge[t];
  if (v == NEG_INF) return;
  int d = dst[t >> 3];
  atomicMax(&menc[d * HH + (t & 7)], fenc(v));
}

// den/num initialized with the self-loop contribution (full write, no zeroing).
__global__ void k_node_init(const float* __restrict__ eself, const unsigned* __restrict__ menc,
                            const float* __restrict__ xl,
                            float* __restrict__ den, float* __restrict__ num) {
  int t = blockIdx.x * blockDim.x + threadIdx.x;
  if (t >= NN * HH) return;
  int n = t >> 3, h = t & 7;
  float ps = __expf(eself[t] - fdec(menc[t]));
  den[t] = ps;
  const float* lp = xl + (size_t)n * DD + h * CC;
  float* np = num + (size_t)n * DD + h * CC;
#pragma unroll
  for (int c = 0; c < CC; ++c) np[c] = ps * lp[c];
}

__global__ void k_edge_accum(const float* __restrict__ eedge, const unsigned* __restrict__ menc,
                             const float* __restrict__ xl,
                             const int* __restrict__ src, const int* __restrict__ dst,
                             float* __restrict__ den, float* __restrict__ num) {
  int t = blockIdx.x * blockDim.x + threadIdx.x;
  if (t >= EE * HH) return;
  float v = eedge[t];
  if (v == NEG_INF) return;
  int e = t >> 3, h = t & 7;
  int s = src[e], d = dst[e];
  float pe = __expf(v - fdec(menc[d * HH + h]));
  atomicAdd(&den[d * HH + h], pe);
  const float* lp = xl + (size_t)s * DD + h * CC;
  float* np = num + (size_t)d * DD + h * CC;
#pragma unroll
  for (int c = 0; c < CC; ++c) atomicAdd(&np[c], pe * lp[c]);
}

__global__ void k_attn_out(const float* __restrict__ num, const float* __restrict__ den,
                           const float* __restrict__ cbias, const int* __restrict__ nmask,
                           float* __restrict__ xc) {
  int t = blockIdx.x * blockDim.x + threadIdx.x;
  if (t >= NN * DD) return;
  int n = t / DD, j = t % DD;
  xc[t] = nmask[n] ? (num[t] / den[n * HH + (j >> 5)] + cbias[j]) : 0.0f;
}

// ------------------------------ GraphNorm -----------------------------------

__global__ __launch_bounds__(256) void k_gstats1(const float* __restrict__ x,
                                                 const int* __restrict__ nmask,
                                                 float* __restrict__ gmean,
                                                 float* __restrict__ gcnt) {
  int g = blockIdx.x, j = threadIdx.x;
  __shared__ float s_cnt;
  if (j == 0) {
    float c = 0.0f;
    for (int i = 0; i < NPG; ++i) c += nmask[g * NPG + i] ? 1.0f : 0.0f;
    s_cnt = fmaxf(c, 1.0f);
    gcnt[g] = s_cnt;
  }
  __syncthreads();
  float s = 0.0f;
  for (int i = 0; i < NPG; ++i) {
    int n = g * NPG + i;
    if (nmask[n]) s += x[(size_t)n * DD + j];
  }
  gmean[g * DD + j] = s / s_cnt;
}

__global__ __launch_bounds__(256) void k_gstats2(const float* __restrict__ x,
                                                 const int* __restrict__ nmask,
                                                 const float* __restrict__ alpha,
                                                 const float* __restrict__ gmean,
                                                 const float* __restrict__ gcnt,
                                                 float* __restrict__ gvar) {
  int g = blockIdx.x, j = threadIdx.x;
  float am = alpha[j] * gmean[g * DD + j];
  float s = 0.0f;
  for (int i = 0; i < NPG; ++i) {
    int n = g * NPG + i;
    if (nmask[n]) { float d = x[(size_t)n * DD + j] - am; s += d * d; }
  }
  gvar[g * DD + j] = s / gcnt[g];
}

__global__ void k_norm_relu(float* __restrict__ x, const int* __restrict__ nmask,
                            const float* __restrict__ alpha, const float* __restrict__ weight,
                            const float* __restrict__ bias,
                            const float* __restrict__ gmean, const float* __restrict__ gvar) {
  int t = blockIdx.x * blockDim.x + threadIdx.x;
  if (t >= NN * DD) return;
  int n = t / DD, j = t % DD, g = n / NPG;
  if (!nmask[n]) { x[t] = 0.0f; return; }
  float sub = x[t] - alpha[j] * gmean[g * DD + j];
  float y = weight[j] * sub * rsqrtf(gvar[g * DD + j] + 1e-5f) + bias[j];
  x[t] = fmaxf(y, 0.0f);
}

// ------------------------------ TopK pooling --------------------------------
// One workgroup (32 wave32s) per graph; stable descending rank by counting.
__global__ __launch_bounds__(1024) void k_topk(float* __restrict__ x, int* __restrict__ nmask,
                                               const float* __restrict__ pw) {
  __shared__ float ssc[NPG];
  __shared__ int sacnt;
  int g = blockIdx.x, i = threadIdx.x, n = g * NPG + i;
  if (i == 0) sacnt = 0;
  float sc = NEG_INF;
  int act = 0;
  if (i < NPG) {
    act = nmask[n];
    if (act) {
      float dot = 0.0f, w2 = 0.0f;
      for (int j = 0; j < DD; ++j) {
        float wv = pw[j];
        dot += x[(size_t)n * DD + j] * wv;
        w2 += wv * wv;
      }
      sc = dot / sqrtf(w2);
    }
    ssc[i] = sc;
  }
  __syncthreads();
  if (act) atomicAdd(&sacnt, 1);
  __syncthreads();
  if (i >= NPG) return;
  int r = 0;
  for (int j = 0; j < NPG; ++j) {
    float o = ssc[j];
    r += (o > sc) || (o == sc && j < i);
  }
  float k = ceilf(0.8f * (float)sacnt);
  int keep = act && ((float)r < k);
  nmask[n] = keep;
  float t = tanhf(sc);
  for (int j = 0; j < DD; ++j)
    x[(size_t)n * DD + j] = keep ? x[(size_t)n * DD + j] * t : 0.0f;
}

__global__ void k_emask_update(int* __restrict__ emask, const int* __restrict__ nmask,
                               const int* __restrict__ src, const int* __restrict__ dst) {
  int e = blockIdx.x * blockDim.x + threadIdx.x;
  if (e >= EE) return;
  emask[e] = emask[e] && nmask[src[e]] && nmask[dst[e]];
}

// --------------------------- readout + MLP head -----------------------------

__global__ __launch_bounds__(256) void k_pool(const float* __restrict__ x,
                                              const int* __restrict__ nmask,
                                              float* __restrict__ hbuf) {
  int g = blockIdx.x, j = threadIdx.x;
  float s = 0.0f, mx = NEG_INF, c = 0.0f;
  for (int i = 0; i < NPG; ++i) {
    int n = g * NPG + i;
    if (nmask[n]) { float v = x[(size_t)n * DD + j]; s += v; mx = fmaxf(mx, v); c += 1.0f; }
  }
  hbuf[g * 2 * DD + j] = s / fmaxf(c, 1.0f);
  hbuf[g * 2 * DD + DD + j] = mx;
}

__global__ __launch_bounds__(256) void k_mlp(const float* __restrict__ hbuf,
                                             const float* __restrict__ W0, const float* __restrict__ b0,
                                             const float* __restrict__ W1, const float* __restrict__ b1,
                                             const float* __restrict__ W2, const float* __restrict__ b2,
                                             float* __restrict__ out) {
  __shared__ float h0[2 * DD];
  __shared__ float h1[DD];
  __shared__ float h2[128];
  __shared__ float z[2];
  int g = blockIdx.x, t = threadIdx.x;
  h0[t] = hbuf[g * 2 * DD + t];
  h0[DD + t] = hbuf[g * 2 * DD + DD + t];
  __syncthreads();
  {
    float a = b0[t];
    for (int i = 0; i < 2 * DD; ++i) a += h0[i] * W0[i * DD + t];
    h1[t] = fmaxf(a, 0.0f);
  }
  __syncthreads();
  if (t < 128) {
    float a = b1[t];
    for (int i = 0; i < DD; ++i) a += h1[i] * W1[i * 128 + t];
    h2[t] = fmaxf(a, 0.0f);
  }
  __syncthreads();
  if (t < 2) {
    float a = b2[t];
    for (int i = 0; i < 128; ++i) a += h2[i] * W2[i * 2 + t];
    z[t] = a;
  }
  __syncthreads();
  if (t < 2) {
    float mz = fmaxf(z[0], z[1]);
    float l = mz + logf(__expf(z[0] - mz) + __expf(z[1] - mz));
    out[g * 2 + t] = z[t] - l;
  }
}

// ------------------------------- launcher -----------------------------------

static inline int cdiv(int a, int b) { return (a + b - 1) / b; }

extern "C" void kernel_launch(void* const* d_in, const int* in_sizes, int n_in,
                              void* d_out, int out_size, void* d_ws, size_t ws_size,
                              hipStream_t stream) {
  const float* X0    = (const float*)d_in[0];
  const float* EATTR = (const float*)d_in[1];
  const float* We[2]    = {(const float*)d_in[2],  (const float*)d_in[9]};
  const float* Wl[2]    = {(const float*)d_in[3],  (const float*)d_in[10]};
  const float* Wr[2]    = {(const float*)d_in[4],  (const float*)d_in[11]};
  const float* att[2]   = {(const float*)d_in[5],  (const float*)d_in[12]};
  const float* cbias[2] = {(const float*)d_in[6],  (const float*)d_in[13]};
  const float* bl[2]    = {(const float*)d_in[7],  (const float*)d_in[14]};
  const float* br[2]    = {(const float*)d_in[8],  (const float*)d_in[15]};
  const float* W0 = (const float*)d_in[16]; const float* b0 = (const float*)d_in[17];
  const float* W1 = (const float*)d_in[18]; const float* b1 = (const float*)d_in[19];
  const float* W2 = (const float*)d_in[20]; const float* b2 = (const float*)d_in[21];
  const float* nalpha[2]  = {(const float*)d_in[22], (const float*)d_in[25]};
  const float* nbias[2]   = {(const float*)d_in[23], (const float*)d_in[26]};
  const float* nweight[2] = {(const float*)d_in[24], (const float*)d_in[27]};
  const float* pw[2] = {(const float*)d_in[28], (const float*)d_in[29]};
  const int* ei = (const int*)d_in[30];
  const int* src = ei;
  const int* dst = ei + EE;

  // workspace carve-up (floats)
  float* ws = (float*)d_ws;
  size_t o = 0;
  float* f_xl    = ws + o; o += (size_t)NN * DD;
  float* f_xr    = ws + o; o += (size_t)NN * DD;
  float* f_num   = ws + o; o += (size_t)NN * DD;
  float* f_xc    = ws + o; o += (size_t)NN * DD;
  float* f_eedge = ws + o; o += (size_t)EE * HH;
  float* f_eself = ws + o; o += (size_t)NN * HH;
  float* f_den   = ws + o; o += (size_t)NN * HH;
  unsigned* u_m  = (unsigned*)(ws + o); o += (size_t)NN * HH;
  float* f_sacc  = ws + o; o += (size_t)NN * FE;   // self_acc then self_cnt contiguous
  float* f_scnt  = ws + o; o += (size_t)NN;
  int* i_nmask   = (int*)(ws + o); o += (size_t)NN;
  int* i_emask   = (int*)(ws + o); o += (size_t)EE;
  float* f_gmean = ws + o; o += (size_t)NG * DD;
  float* f_gvar  = ws + o; o += (size_t)NG * DD;
  float* f_gcnt  = ws + o; o += (size_t)NG;
  float* f_hbuf  = ws + o; o += (size_t)NG * 2 * DD;

  const int TB = 256;
  k_init_masks<<<cdiv(EE, TB), TB, 0, stream>>>(i_nmask, i_emask);

  const float* xin = X0;
  int K = 128;
  for (int l = 0; l < 2; ++l) {
    // zero self-attr accumulators (sacc + scnt contiguous)
    k_zero<<<cdiv(NN * (FE + 1), TB), TB, 0, stream>>>(f_sacc, NN * (FE + 1));
    // node transforms via WMMA fp32 GEMM:
    // grid = (row-tile groups of 8) x (column tiles); B tile staged in LDS
    dim3 ggrid(cdiv(NN / 16, 8), DD / 16);
    k_wmma_gemm_bias<<<ggrid, TB, 0, stream>>>(xin, Wl[l], bl[l], f_xl, NN, K, DD);
    k_wmma_gemm_bias<<<ggrid, TB, 0, stream>>>(xin, Wr[l], br[l], f_xr, NN, K, DD);
    // GATv2 attention (scatter formulation)
    k_selfattr<<<cdiv(EE, TB), TB, 0, stream>>>(EATTR, src, dst, i_emask, f_sacc, f_scnt);
    k_edge_scores<<<cdiv(EE * HH, TB), TB, 0, stream>>>(f_xl, f_xr, EATTR, We[l], att[l],
                                                        src, dst, i_emask, f_eedge);
    k_self_scores<<<cdiv(NN * HH, TB), TB, 0, stream>>>(f_xl, f_xr, We[l], att[l],
                                                        f_sacc, f_scnt, f_eself, u_m);
    k_edge_max<<<cdiv(EE * HH, TB), TB, 0, stream>>>(f_eedge, dst, u_m);
    k_node_init<<<cdiv(NN * HH, TB), TB, 0, stream>>>(f_eself, u_m, f_xl, f_den, f_num);
    k_edge_accum<<<cdiv(EE * HH, TB), TB, 0, stream>>>(f_eedge, u_m, f_xl, src, dst,
                                                       f_den, f_num);
    k_attn_out<<<cdiv(NN * DD, TB), TB, 0, stream>>>(f_num, f_den, cbias[l], i_nmask, f_xc);
    // GraphNorm + ReLU
    k_gstats1<<<NG, TB, 0, stream>>>(f_xc, i_nmask, f_gmean, f_gcnt);
    k_gstats2<<<NG, TB, 0, stream>>>(f_xc, i_nmask, nalpha[l], f_gmean, f_gcnt, f_gvar);
    k_norm_relu<<<cdiv(NN * DD, TB), TB, 0, stream>>>(f_xc, i_nmask, nalpha[l], nweight[l],
                                                      nbias[l], f_gmean, f_gvar);
    // TopK pooling
    k_topk<<<NG, 1024, 0, stream>>>(f_xc, i_nmask, pw[l]);
    k_emask_update<<<cdiv(EE, TB), TB, 0, stream>>>(i_emask, i_nmask, src, dst);
    xin = f_xc;
    K = DD;
  }

  k_pool<<<NG, TB, 0, stream>>>(f_xc, i_nmask, f_hbuf);
  k_mlp<<<NG, TB, 0, stream>>>(f_hbuf, W0, b0, W1, b1, W2, b2, (float*)d_out);
}